// AMPSShare_22333829939477
// MI455X (gfx1250) — compile-verified
//
#include <hip/hip_runtime.h>

// MPS autoregressive log-prob scan, fp32 WMMA formulation for gfx1250.
// Two independent 16-sample chains per wave (ILP-2 on the latency-bound scan),
// one shared B operand per site:
//   D(16x16) = A(16x4) x B(4x16),  B = [T0 | T1 | t[:,0,:] | 0]
// Branchless inner loop (EXEC stays all-ones), raw v_exp/v_log softmax.

#define N_SITES 784
#define BS      16384

typedef __attribute__((ext_vector_type(2))) float v2f;
typedef __attribute__((ext_vector_type(8))) float v8f;

__launch_bounds__(256)
__global__ void amps_mps_logprob(const float* __restrict__ data,
                                 const float* __restrict__ tensors,
                                 float* __restrict__ out)
{
    __shared__ float smem[8 * 2 * 256];        // 8 waves * 2 chains * 16x16 f32
    const int lane = threadIdx.x & 31;         // wave32
    const int wave = threadIdx.x >> 5;
    const int half = lane >> 4;                // A-operand K-split half
    const int m    = lane & 15;                // sample-in-tile == B column
    const int tileBase = (blockIdx.x * 8 + wave) * 32;   // 32 samples per wave
    float* wt0 = smem + wave * 512;
    float* wt1 = wt0 + 256;

    // Loop-invariant B offsets into the 32-float per-site block, idx(l,r,p)=l*8+r*2+p.
    // B layout (4x16 f32): lane n, half h holds B[2h+0,n] (v0) and B[2h+1,n] (v1).
    const int kk = half * 2;
    int   off0  = 0;
    float bmask = 1.0f;                        // branchless zero for cols 10-15
    if      (m < 4)  off0 = kk * 8 + m * 2;            // T0 col m        (p=0)
    else if (m < 8)  off0 = kk * 8 + (m - 4) * 2 + 1;  // T1 col m-4      (p=1)
    else if (m == 8) off0 = kk * 8;                    // t[:,0,0]
    else if (m == 9) off0 = kk * 8 + 1;                // t[:,0,1]
    else             bmask = 0.0f;

    const size_t drow0 = (size_t)(tileBase +      m) * N_SITES;
    const size_t drow1 = (size_t)(tileBase + 16 + m) * N_SITES;

    constexpr float LOG2E = 1.44269504088896340736f;
    constexpr float LN2   = 0.69314718055994530942f;

    // ---- site 0: left0[r] = tensors[0,0,r,p], lx0 = log_softmax(tensors[0,0,0,:])
    const float z0 = tensors[0], z1 = tensors[1];
    const float mx0  = fmaxf(z0, z1);
    const float lse0 = mx0 + __builtin_amdgcn_logf(
        __builtin_amdgcn_exp2f((z0 - mx0) * LOG2E) +
        __builtin_amdgcn_exp2f((z1 - mx0) * LOG2E)) * LN2;

    const float x0a = data[drow0];
    const float x0b = data[drow1];
    const int pa = (x0a > 0.5f) ? 0 : 1;
    const int pb = (x0b > 0.5f) ? 0 : 1;
    v2f a0, a1;
    a0.x = tensors[(2 * half + 0) * 2 + pa];   // A: half0->K0,K1 ; half1->K2,K3
    a0.y = tensors[(2 * half + 1) * 2 + pa];
    a1.x = tensors[(2 * half + 0) * 2 + pb];
    a1.y = tensors[(2 * half + 1) * 2 + pb];
    float lp0 = ((x0a > 0.5f) ? z0 : z1) - lse0;
    float lp1 = ((x0b > 0.5f) ? z0 : z1) - lse0;

    const float* tp = tensors + 32;
    for (int site = 1; site < N_SITES; ++site, tp += 32) {
        // Shared B operand for both chains (L2-resident, branchless masking).
        v2f b;
        b.x = bmask * tp[off0];
        b.y = bmask * tp[off0 + 8];
        const float xv0 = data[drow0 + site];              // HBM stream
        const float xv1 = data[drow1 + site];

        v8f c = {};
        // 8 args: (neg_a, A, neg_b, B, c_mod, C, reuse_a, reuse_b)
        v8f d0 = __builtin_amdgcn_wmma_f32_16x16x4_f32(
            false, a0, false, b, (short)0, c, false, false);
        v8f d1 = __builtin_amdgcn_wmma_f32_16x16x4_f32(
            false, a1, false, b, (short)0, c, false, false);

        // Scatter D tiles to LDS: VGPR r = row (r + 8*half), col m.
        #pragma unroll
        for (int r = 0; r < 8; ++r) {
            wt0[(r + 8 * half) * 16 + m] = d0[r];
            wt1[(r + 8 * half) * 16 + m] = d1[r];
        }
        asm volatile("s_wait_dscnt 0" ::: "memory");        // same-wave DS fence

        // Gather new left in A layout (row m, col block chosen by data bit).
        const int base0 = (xv0 > 0.5f) ? 0 : 4;
        const int base1 = (xv1 > 0.5f) ? 0 : 4;
        const float2 nl0 = *(const float2*)(wt0 + m * 16 + base0 + 2 * half);
        const float2 nl1 = *(const float2*)(wt1 + m * 16 + base1 + 2 * half);
        a0.x = nl0.x; a0.y = nl0.y;
        a1.x = nl1.x; a1.y = nl1.y;

        // Log-softmax on ALL lanes (halves read identical logits; no divergence).
        const float2 lz0 = *(const float2*)(wt0 + m * 16 + 8);
        const float2 lz1 = *(const float2*)(wt1 + m * 16 + 8);
        {
            const float mm = fmaxf(lz0.x, lz0.y);
            const float s  = __builtin_amdgcn_exp2f((lz0.x - mm) * LOG2E)
                           + __builtin_amdgcn_exp2f((lz0.y - mm) * LOG2E);
            const float ls = mm + __builtin_amdgcn_logf(s) * LN2;
            lp0 += ((xv0 > 0.5f) ? lz0.x : lz0.y) - ls;
        }
        {
            const float mm = fmaxf(lz1.x, lz1.y);
            const float s  = __builtin_amdgcn_exp2f((lz1.x - mm) * LOG2E)
                           + __builtin_amdgcn_exp2f((lz1.y - mm) * LOG2E);
            const float ls = mm + __builtin_amdgcn_logf(s) * LN2;
            lp1 += ((xv1 > 0.5f) ? lz1.x : lz1.y) - ls;
        }
    }

    if (half == 0) {                           // lanes 0-15 own the results
        out[tileBase +      m] = lp0;
        out[tileBase + 16 + m] = lp1;
    }
}

extern "C" void kernel_launch(void* const* d_in, const int* in_sizes, int n_in,
                              void* d_out, int out_size, void* d_ws, size_t ws_size,
                              hipStream_t stream) {
    const float* data    = (const float*)d_in[0];   // (BS, 784) f32 in {0,1}
    const float* tensors = (const float*)d_in[1];   // (784, 4, 4, 2) f32
    float* out = (float*)d_out;                     // (BS,) f32

    dim3 block(256);                  // 8 waves * 32 samples = 256 samples/block
    dim3 grid(BS / 256);              // 64 blocks, 512 waves, ILP-2 per wave
    amps_mps_logprob<<<grid, block, 0, stream>>>(data, tensors, out);
}